// TTM_linear_30365418783369
// MI455X (gfx1250) — compile-verified
//
#include <hip/hip_runtime.h>
#include <hip/hip_bf16.h>

typedef __attribute__((ext_vector_type(2))) float v2f;
typedef __attribute__((ext_vector_type(8))) float v8f;

#define BATCH   16384
#define RNK     20
#define NCOLS   4096   // 8*8*8*8

// ---------------------------------------------------------------------------
// Stage 1: out0 = x0 (B,20) @ G0 (20,20)  -> A buffer (B,20) row-major
// ---------------------------------------------------------------------------
__global__ void __launch_bounds__(256) k_out0(const float* __restrict__ x0,
                                              const float* __restrict__ G0,
                                              float* __restrict__ A) {
    int tid = blockIdx.x * blockDim.x + threadIdx.x;
    if (tid >= BATCH * RNK) return;
    int b = tid / RNK, s = tid % RNK;
    float acc = 0.f;
    #pragma unroll
    for (int r = 0; r < RNK; ++r) acc += x0[b * RNK + r] * G0[r * RNK + s];
    A[tid] = acc;
}

// ---------------------------------------------------------------------------
// Stage 2: output TT cores
//   o_i[s][n][t] = sum_{r,m} x_i[r][m][s] * G_i[r][m][n][t]   (i=1..3, 3200 ea)
//   o4[t][n]     = sum_{r,m} x4[r][m]     * G4[r][m][n][t]    (160)
// ---------------------------------------------------------------------------
__global__ void __launch_bounds__(256) k_cores(const float* __restrict__ x1,
                                               const float* __restrict__ x2,
                                               const float* __restrict__ x3,
                                               const float* __restrict__ x4,
                                               const float* __restrict__ G1,
                                               const float* __restrict__ G2,
                                               const float* __restrict__ G3,
                                               const float* __restrict__ G4,
                                               float* __restrict__ o1,
                                               float* __restrict__ o2,
                                               float* __restrict__ o3,
                                               float* __restrict__ o4) {
    int tid = blockIdx.x * blockDim.x + threadIdx.x;
    if (tid < 3 * 3200) {
        int which = tid / 3200;
        int idx   = tid % 3200;
        const float* x = (which == 0) ? x1 : (which == 1) ? x2 : x3;
        const float* G = (which == 0) ? G1 : (which == 1) ? G2 : G3;
        float*       o = (which == 0) ? o1 : (which == 1) ? o2 : o3;
        int s = idx / 160, rem = idx % 160, n = rem / RNK, t = rem % RNK;
        float acc = 0.f;
        for (int rm = 0; rm < 160; ++rm)            // rm = r*8+m
            acc += x[rm * RNK + s] * G[(rm * 8 + n) * RNK + t];
        o[idx] = acc;
    } else if (tid < 3 * 3200 + 160) {
        int idx = tid - 3 * 3200;                   // idx = t*8+n
        int t = idx / 8, n = idx % 8;
        float acc = 0.f;
        for (int rm = 0; rm < 160; ++rm)
            acc += x4[rm] * G4[(rm * 8 + n) * RNK + t];
        o4[idx] = acc;
    }
}

// ---------------------------------------------------------------------------
// Stage 3: c12[s][n1][n2][u] = sum_t o1[s][n1][t] * o2[t][n2][u]   (25600)
// ---------------------------------------------------------------------------
__global__ void __launch_bounds__(256) k_chain12(const float* __restrict__ o1,
                                                 const float* __restrict__ o2,
                                                 float* __restrict__ c12) {
    int tid = blockIdx.x * blockDim.x + threadIdx.x;
    if (tid >= RNK * 8 * 8 * RNK) return;
    int u = tid % RNK; int r = tid / RNK;
    int n2 = r % 8; r /= 8;
    int n1 = r % 8; int s = r / 8;
    float acc = 0.f;
    #pragma unroll
    for (int t = 0; t < RNK; ++t)
        acc += o1[s * 160 + n1 * RNK + t] * o2[t * 160 + n2 * RNK + u];
    c12[tid] = acc;
}

// ---------------------------------------------------------------------------
// Stage 4: c123[s][n1][n2][n3][v] = sum_t c12[s][n1][n2][t] * o3[t][n3][v] (204800)
// ---------------------------------------------------------------------------
__global__ void __launch_bounds__(256) k_chain123(const float* __restrict__ c12,
                                                  const float* __restrict__ o3,
                                                  float* __restrict__ c123) {
    int tid = blockIdx.x * blockDim.x + threadIdx.x;
    if (tid >= RNK * 8 * 8 * 8 * RNK) return;
    int v = tid % RNK; int r = tid / RNK;
    int n3 = r % 8; r /= 8;
    int n2 = r % 8; r /= 8;
    int n1 = r % 8; int s = r / 8;
    float acc = 0.f;
    #pragma unroll
    for (int t = 0; t < RNK; ++t)
        acc += c12[((s * 8 + n1) * 8 + n2) * RNK + t] * o3[t * 160 + n3 * RNK + v];
    c123[tid] = acc;
}

// ---------------------------------------------------------------------------
// Stage 5: W[s][col] = sum_v c123[s][pre][v] * o4[v][n4]
//   col = n1*512 + n2*64 + n3*8 + n4, pre = col/8.  W row-major (20, 4096)
// ---------------------------------------------------------------------------
__global__ void __launch_bounds__(256) k_wmat(const float* __restrict__ c123,
                                              const float* __restrict__ o4,
                                              float* __restrict__ W) {
    int tid = blockIdx.x * blockDim.x + threadIdx.x;
    if (tid >= RNK * NCOLS) return;
    int col = tid % NCOLS, s = tid / NCOLS;
    int n4 = col % 8, pre = col / 8;                // pre = n1*64 + n2*8 + n3
    float acc = 0.f;
    #pragma unroll
    for (int v = 0; v < RNK; ++v)
        acc += c123[(s * 512 + pre) * RNK + v] * o4[v * 8 + n4];
    W[tid] = acc;
}

// ---------------------------------------------------------------------------
// Stage 6 (the only HBM-heavy stage, ~256 MiB stores):
//   out (16384,4096) = A (16384,20) x W (20,4096) via V_WMMA_F32_16X16X4_F32.
//   One 16x16 output tile per wave, K=20 -> 5 wmma steps of k=4.
//   ISA 32-bit layouts: A: a[v] = A[row=ln][k = v + 2*half]
//                       B: b[v] = W[k = v + 2*half][col=ln]
//                       D: c[v] = out[M = v + 8*half][N = ln]
// ---------------------------------------------------------------------------
__global__ void __launch_bounds__(256) k_gemm_wmma(const float* __restrict__ A,
                                                   const float* __restrict__ W,
                                                   float* __restrict__ out) {
    const int lane = threadIdx.x & 31;
    const int wave = threadIdx.x >> 5;
    const int half = lane >> 4;           // 0: lanes 0-15, 1: lanes 16-31
    const int ln   = lane & 15;

    const unsigned tileId = blockIdx.x * 8u + (unsigned)wave;
    const unsigned nt = tileId & 255u;    // 256 N-tiles (fast: waves in a block
    const unsigned mt = tileId >> 8;      //  share M rows, write adjacent cols)

    // A row for this lane: stride 20 floats (80B -> float2-aligned at even k)
    const float* arow = A + (mt * 16u + (unsigned)ln) * RNK + 2 * half;
    // B column for this lane, pre-offset by the half's k-base row
    const float* bcol = W + (unsigned)(2 * half) * NCOLS + nt * 16u + (unsigned)ln;

    v8f c = {};
    #pragma unroll
    for (int kk = 0; kk < 5; ++kk) {      // k = 4*kk + v + 2*half covers 0..19
        v2f a;
        {   // aligned 8-byte load of (k, k+1) for this half
            const v2f av = *(const v2f*)(arow + 4 * kk);
            a = av;
        }
        v2f b;
        b.x = bcol[(4 * kk + 0) * NCOLS];
        b.y = bcol[(4 * kk + 1) * NCOLS];
        c = __builtin_amdgcn_wmma_f32_16x16x4_f32(
                /*neg_a=*/false, a, /*neg_b=*/false, b,
                /*c_mod=*/(short)0, c, /*reuse_a=*/false, /*reuse_b=*/false);
    }

    float* orow = out + (mt * 16u + (unsigned)(8 * half)) * NCOLS + nt * 16u + (unsigned)ln;
    #pragma unroll
    for (int v = 0; v < 8; ++v)
        orow[v * NCOLS] = c[v];
}

// ---------------------------------------------------------------------------
// Host-side launcher
// ---------------------------------------------------------------------------
extern "C" void kernel_launch(void* const* d_in, const int* in_sizes, int n_in,
                              void* d_out, int out_size, void* d_ws, size_t ws_size,
                              hipStream_t stream) {
    const float* x0 = (const float*)d_in[0];
    const float* x1 = (const float*)d_in[1];
    const float* x2 = (const float*)d_in[2];
    const float* x3 = (const float*)d_in[3];
    const float* x4 = (const float*)d_in[4];
    const float* G0 = (const float*)d_in[5];
    const float* G1 = (const float*)d_in[6];
    const float* G2 = (const float*)d_in[7];
    const float* G3 = (const float*)d_in[8];
    const float* G4 = (const float*)d_in[9];
    float* out = (float*)d_out;

    // workspace layout (floats)
    float* ws   = (float*)d_ws;
    float* A    = ws;                       // 16384*20 = 327680
    float* o1   = A    + BATCH * RNK;       // 3200
    float* o2   = o1   + 3200;              // 3200
    float* o3   = o2   + 3200;              // 3200
    float* o4   = o3   + 3200;              // 160
    float* c12  = o4   + 160;               // 25600
    float* c123 = c12  + 25600;             // 204800
    float* W    = c123 + 204800;            // 81920   (total ~2.6 MB)

    // Stage 1: out0 = x0 @ G0
    k_out0<<<(BATCH * RNK + 255) / 256, 256, 0, stream>>>(x0, G0, A);
    // Stage 2: output cores o1..o4
    k_cores<<<(3 * 3200 + 160 + 255) / 256, 256, 0, stream>>>(
        x1, x2, x3, x4, G1, G2, G3, G4, o1, o2, o3, o4);
    // Stage 3-5: chain contraction -> W (20, 4096)
    k_chain12 <<<(RNK * 8 * 8 * RNK        + 255) / 256, 256, 0, stream>>>(o1, o2, c12);
    k_chain123<<<(RNK * 8 * 8 * 8 * RNK    + 255) / 256, 256, 0, stream>>>(c12, o3, c123);
    k_wmat    <<<(RNK * NCOLS              + 255) / 256, 256, 0, stream>>>(c123, o4, W);
    // Stage 6: big WMMA GEMM, 1024x256 tiles, 8 waves (tiles) per block
    const int total_tiles = (BATCH / 16) * (NCOLS / 16);  // 262144
    k_gemm_wmma<<<total_tiles / 8, 256, 0, stream>>>(A, W, out);
}